// GAT7Model_6124623364715
// MI455X (gfx1250) — compile-verified
//
#include <hip/hip_runtime.h>
#include <hip/hip_bf16.h>

// ---------------------------------------------------------------------------
// GATv2 x4 + MLP head for MI455X (gfx1250, wave32).
// Dense node transforms use V_WMMA_F32_16X16X4_F32; edge phase is
// bandwidth/atomics bound and uses float4 gathers + f32 atomics.
// ---------------------------------------------------------------------------

typedef __attribute__((ext_vector_type(2))) float v2f;
typedef __attribute__((ext_vector_type(8))) float v8f;

#define CH 64  // hidden/output channel count for every GAT layer

// ---------------------------------------------------------------------------
// Node linear: Y[m, n] = sum_k X[m,k] * W[k,n] + b[n]
// One wave (32 threads) computes a 16x16 tile with WMMA f32 16x16x4.
//   A 16x4 layout : lanes 0-15 hold rows M=lane (K=k0,k0+1 in the 2 VGPRs),
//                   lanes 16-31 hold rows M=lane-16 (K=k0+2,k0+3).
//   B 4x16 layout : mirrored (cols across lanes, 2 K-rows per lane).
//   C/D          : VGPR v, lane half h -> (M = v + 8h, N = lane&15).
// ---------------------------------------------------------------------------
__global__ __launch_bounds__(32)
void node_linear_wmma(const float* __restrict__ X,  // [N, din]
                      const float* __restrict__ W,  // [din, CH]
                      const float* __restrict__ b,  // [CH]
                      float* __restrict__ Y,        // [N, CH]
                      int din) {
  const int lane = threadIdx.x;        // 0..31, full wave => EXEC all ones
  const int m0   = blockIdx.x * 16;
  const int n0   = blockIdx.y * 16;
  const int idx  = lane & 15;          // row for A, col for B/D
  const int half = lane >> 4;          // 0 or 1
  const int koff = half * 2;

  v8f acc = {};
  const float* arow = X + (size_t)(m0 + idx) * din;
  for (int k = 0; k < din; k += 4) {
    v2f a, bm;
    a.x = arow[k + koff];
    a.y = arow[k + koff + 1];
    const float* bp = W + (size_t)(k + koff) * CH + (n0 + idx);
    bm.x = bp[0];
    bm.y = bp[CH];
    acc = __builtin_amdgcn_wmma_f32_16x16x4_f32(
        /*neg_a=*/false, a, /*neg_b=*/false, bm,
        /*c_mod=*/(short)0, acc, /*reuse_a=*/false, /*reuse_b=*/false);
  }

  const float bias = b[n0 + idx];
#pragma unroll
  for (int v = 0; v < 8; ++v) {
    Y[(size_t)(m0 + v + 8 * half) * CH + (n0 + idx)] = acc[v] + bias;
  }
}

// ---------------------------------------------------------------------------
// Monotonic uint key for float atomic-max over all finite values (handles
// negatives): f >= g  <=>  key(f) >= key(g).
// ---------------------------------------------------------------------------
__device__ __forceinline__ unsigned f2ord(float f) {
  unsigned u = __float_as_uint(f);
  return (u & 0x80000000u) ? ~u : (u | 0x80000000u);
}
__device__ __forceinline__ float ord2f(unsigned k) {
  unsigned u = (k & 0x80000000u) ? (k ^ 0x80000000u) : ~k;
  return __uint_as_float(u);
}

// ---------------------------------------------------------------------------
// Edge pass 1: score[e] = att . leaky_relu(xl[src] + xr[dst] + ea*We, 0.2)
// plus running segment max into m_ord[dst].
// ---------------------------------------------------------------------------
__global__ void edge_score_kernel(const float* __restrict__ xl,
                                  const float* __restrict__ xr,
                                  const int* __restrict__ src,
                                  const int* __restrict__ dst,
                                  const float* __restrict__ eattr,  // [E]
                                  const float* __restrict__ We,     // [CH]
                                  const float* __restrict__ att,    // [CH]
                                  float* __restrict__ score,
                                  unsigned* __restrict__ m_ord,
                                  int E) {
  int e = blockIdx.x * blockDim.x + threadIdx.x;
  if (e >= E) return;
  const int s = src[e];
  const int d = dst[e];
  const float ea = eattr[e];
  const float4* pl = (const float4*)(xl + (size_t)s * CH);
  const float4* pr = (const float4*)(xr + (size_t)d * CH);
  const float4* pw = (const float4*)We;
  const float4* pa = (const float4*)att;
  float acc = 0.0f;
#pragma unroll
  for (int c = 0; c < CH / 4; ++c) {
    float4 l = pl[c], r = pr[c], w = pw[c], a = pa[c];
    float v0 = l.x + r.x + ea * w.x;
    float v1 = l.y + r.y + ea * w.y;
    float v2 = l.z + r.z + ea * w.z;
    float v3 = l.w + r.w + ea * w.w;
    v0 = v0 > 0.0f ? v0 : 0.2f * v0;
    v1 = v1 > 0.0f ? v1 : 0.2f * v1;
    v2 = v2 > 0.0f ? v2 : 0.2f * v2;
    v3 = v3 > 0.0f ? v3 : 0.2f * v3;
    acc += v0 * a.x + v1 * a.y + v2 * a.z + v3 * a.w;
  }
  score[e] = acc;
  atomicMax(&m_ord[d], f2ord(acc));
}

// ---------------------------------------------------------------------------
// Edge pass 2: a[e] = exp(score[e] - m[dst]);  denom[dst] += a[e]
// ---------------------------------------------------------------------------
__global__ void edge_exp_kernel(const float* __restrict__ score,
                                const int* __restrict__ dst,
                                const unsigned* __restrict__ m_ord,
                                float* __restrict__ aexp,
                                float* __restrict__ denom,
                                int E) {
  int e = blockIdx.x * blockDim.x + threadIdx.x;
  if (e >= E) return;
  const int d = dst[e];
  const float m = ord2f(m_ord[d]);
  const float a = expf(score[e] - m);
  aexp[e] = a;
  atomicAdd(&denom[d], a);
}

// ---------------------------------------------------------------------------
// Edge pass 3: out[dst, c] += xl[src, c] * a[e] / (denom[dst] + eps)
// One thread per (edge, channel) -> 64 consecutive threads share one edge
// (coalesced 256B row reads of xl, coalesced atomics on out rows).
// ---------------------------------------------------------------------------
__global__ void edge_aggregate_kernel(const float* __restrict__ xl,
                                      const int* __restrict__ src,
                                      const int* __restrict__ dst,
                                      const float* __restrict__ aexp,
                                      const float* __restrict__ denom,
                                      float* __restrict__ out,
                                      long long total) {
  long long idx = (long long)blockIdx.x * blockDim.x + threadIdx.x;
  if (idx >= total) return;
  const int e = (int)(idx >> 6);
  const int c = (int)(idx & 63);
  const int d = dst[e];
  const float alpha = aexp[e] / (denom[d] + 1e-16f);
  atomicAdd(&out[(size_t)d * CH + c], xl[(size_t)src[e] * CH + c] * alpha);
}

// ---------------------------------------------------------------------------
// x[i,c] = relu(x[i,c] + bo[c])   (in place on the edge accumulator)
// ---------------------------------------------------------------------------
__global__ void bias_relu_kernel(float* __restrict__ x,
                                 const float* __restrict__ bo,
                                 long long total) {
  long long idx = (long long)blockIdx.x * blockDim.x + threadIdx.x;
  if (idx >= total) return;
  const float v = x[idx] + bo[idx & 63];
  x[idx] = v > 0.0f ? v : 0.0f;
}

// ---------------------------------------------------------------------------
// Readout partial: sum[c] += x[i,c]
// ---------------------------------------------------------------------------
__global__ void readout_sum_kernel(const float* __restrict__ x,
                                   float* __restrict__ sum,
                                   long long total) {
  long long idx = (long long)blockIdx.x * blockDim.x + threadIdx.x;
  if (idx >= total) return;
  atomicAdd(&sum[idx & 63], x[idx]);
}

// ---------------------------------------------------------------------------
// Final MLP head: mean -> 32 -> 16 -> 1 (single block)
// ---------------------------------------------------------------------------
__global__ __launch_bounds__(64)
void final_mlp_kernel(const float* __restrict__ sum, float invN,
                      const float* __restrict__ Wm1, const float* __restrict__ bm1,
                      const float* __restrict__ Wm2, const float* __restrict__ bm2,
                      const float* __restrict__ Wm3, const float* __restrict__ bm3,
                      float* __restrict__ out) {
  __shared__ float mean[64];
  __shared__ float h1[32];
  __shared__ float h2[16];
  const int t = threadIdx.x;
  mean[t] = sum[t] * invN;
  __syncthreads();
  if (t < 32) {
    float a = bm1[t];
    for (int i = 0; i < 64; ++i) a += mean[i] * Wm1[i * 32 + t];
    h1[t] = a > 0.0f ? a : 0.0f;
  }
  __syncthreads();
  if (t < 16) {
    float a = bm2[t];
    for (int i = 0; i < 32; ++i) a += h1[i] * Wm2[i * 16 + t];
    h2[t] = a > 0.0f ? a : 0.0f;
  }
  __syncthreads();
  if (t == 0) {
    float a = bm3[0];
    for (int i = 0; i < 16; ++i) a += h2[i] * Wm3[i];
    out[0] = a;
  }
}

// ---------------------------------------------------------------------------
// Host-side orchestration.
// Input order: 0 features[N,36], 1 edge_attr[E], 2 edge_index[2,E],
//   then per layer l=1..4: Wl, bl, Wr, br, We, att, bo  (indices 3 + 7*(l-1)),
//   then Wm1, bm1, Wm2, bm2, Wm3, bm3 (indices 31..36).
// ---------------------------------------------------------------------------
extern "C" void kernel_launch(void* const* d_in, const int* in_sizes, int n_in,
                              void* d_out, int out_size, void* d_ws, size_t ws_size,
                              hipStream_t stream) {
  const int C_IN = 36;
  const int E = in_sizes[1];          // edge_attr has E elements
  const int N = in_sizes[0] / C_IN;   // 50000

  const float* features  = (const float*)d_in[0];
  const float* edge_attr = (const float*)d_in[1];
  const int*   edge_idx  = (const int*)d_in[2];
  const int*   src = edge_idx;
  const int*   dst = edge_idx + E;

  // Workspace carve-up (floats)
  float* ws = (float*)d_ws;
  size_t ofs = 0;
  float* xl    = ws + ofs; ofs += (size_t)N * CH;
  float* xr    = ws + ofs; ofs += (size_t)N * CH;
  float* bufA  = ws + ofs; ofs += (size_t)N * CH;
  float* bufB  = ws + ofs; ofs += (size_t)N * CH;
  float* score = ws + ofs; ofs += (size_t)E;
  float* aexp  = ws + ofs; ofs += (size_t)E;
  float* denom = ws + ofs; ofs += (size_t)N;
  unsigned* m_ord = (unsigned*)(ws + ofs); ofs += (size_t)N;
  float* sum64 = ws + ofs; ofs += 64;

  const long long nodeTot = (long long)N * CH;
  const long long edgeTot = (long long)E * CH;
  const int BT = 256;
  const dim3 gemmGrid((unsigned)(N / 16), CH / 16);  // N = 50000 = 3125*16

  const float* xcur = features;
  int din = C_IN;
  float* outbuf = bufA;

  for (int l = 0; l < 4; ++l) {
    const int base = 3 + l * 7;
    const float* Wl  = (const float*)d_in[base + 0];
    const float* bl  = (const float*)d_in[base + 1];
    const float* Wr  = (const float*)d_in[base + 2];
    const float* br  = (const float*)d_in[base + 3];
    const float* We  = (const float*)d_in[base + 4];
    const float* att = (const float*)d_in[base + 5];
    const float* bo  = (const float*)d_in[base + 6];

    // xl = x @ Wl + bl ; xr = x @ Wr + br   (WMMA f32)
    node_linear_wmma<<<gemmGrid, 32, 0, stream>>>(xcur, Wl, bl, xl, din);
    node_linear_wmma<<<gemmGrid, 32, 0, stream>>>(xcur, Wr, br, xr, din);

    // Zero accumulators for this layer
    hipMemsetAsync(outbuf, 0, (size_t)nodeTot * sizeof(float), stream);
    hipMemsetAsync(denom, 0, (size_t)N * sizeof(float), stream);
    hipMemsetAsync(m_ord, 0, (size_t)N * sizeof(unsigned), stream);

    // Edge phase
    edge_score_kernel<<<(E + BT - 1) / BT, BT, 0, stream>>>(
        xl, xr, src, dst, edge_attr, We, att, score, m_ord, E);
    edge_exp_kernel<<<(E + BT - 1) / BT, BT, 0, stream>>>(
        score, dst, m_ord, aexp, denom, E);
    edge_aggregate_kernel<<<(unsigned)((edgeTot + BT - 1) / BT), BT, 0, stream>>>(
        xl, src, dst, aexp, denom, outbuf, edgeTot);

    // out = relu(out + bo), becomes next layer's x
    bias_relu_kernel<<<(unsigned)((nodeTot + BT - 1) / BT), BT, 0, stream>>>(
        outbuf, bo, nodeTot);

    xcur = outbuf;
    din = CH;
    outbuf = (outbuf == bufA) ? bufB : bufA;
  }

  // Graph readout + MLP head
  const float* Wm1 = (const float*)d_in[31];
  const float* bm1 = (const float*)d_in[32];
  const float* Wm2 = (const float*)d_in[33];
  const float* bm2 = (const float*)d_in[34];
  const float* Wm3 = (const float*)d_in[35];
  const float* bm3 = (const float*)d_in[36];

  hipMemsetAsync(sum64, 0, 64 * sizeof(float), stream);
  readout_sum_kernel<<<(unsigned)((nodeTot + BT - 1) / BT), BT, 0, stream>>>(
      xcur, sum64, nodeTot);
  final_mlp_kernel<<<1, 64, 0, stream>>>(
      sum64, 1.0f / (float)N, Wm1, bm1, Wm2, bm2, Wm3, bm3, (float*)d_out);
}